// UniGATConv_2594160246976
// MI455X (gfx1250) — compile-verified
//
#include <hip/hip_runtime.h>
#include <math.h>
#include <stdint.h>

// ---- problem constants (from reference) ----
#define Hh 8
#define Cc 32
#define HC 256          // H*C
#define IN_DIM 256
#define NEG_SLOPE 0.2f
#define E_CAP 16384     // workspace cap for number of hyperedges (actual E=10000, read on device)

typedef __attribute__((ext_vector_type(2))) float v2f;
typedef __attribute__((ext_vector_type(8))) float v8f;

// ---------------------------------------------------------------------------
// Stage 0: fill helper (deterministic init of workspace / output each call)
// ---------------------------------------------------------------------------
__global__ void fill_f32(float* __restrict__ p, float val, size_t n) {
    size_t i = (size_t)blockIdx.x * blockDim.x + threadIdx.x;
    if (i < n) p[i] = val;
}

// ---------------------------------------------------------------------------
// Stage 1: X0 = X @ W.T via V_WMMA_F32_16X16X4_F32 with async-LDS A staging.
//
// One block (8 waves) per 16-row M panel. The A panel (16x256 f32 = 16KB) is
// DMA'd into LDS once via GLOBAL_LOAD_ASYNC_TO_LDS_B128 (ASYNCcnt), then each
// wave computes two 16x16 output tiles (n-tiles 2w and 2w+1), reading A
// fragments from LDS (ds_load_b64) and B fragments from W in global (L2-hot).
//
// A 16x4 fragment layout: lane l<16 VGPR0 = A[M=l][K=0]; lanes>=16 K=2;
// VGPR1 = K=1/K=3 -> per-lane float2 at (row l15, k + 2*hi).
// B 4x16 mirrors it (B = W.T, so per-lane float2 at W[(n0+l15)*256 + k+2*hi]).
// D layout: VGPR v -> row m0 + v + 8*hi, col n0 + l15.
// ---------------------------------------------------------------------------
__global__ __launch_bounds__(256) void gemm_wmma(const float* __restrict__ X,
                                                 const float* __restrict__ W,
                                                 float* __restrict__ X0) {
    __shared__ float sA[16 * IN_DIM];           // 16 KB A panel
    const int m0 = blockIdx.x << 4;
    const int t  = threadIdx.x;

    // ---- async DMA: global X panel -> LDS (4 x b128 chunks per thread) ----
    const uint32_t lds_base = (uint32_t)(size_t)&sA[0];   // low 32 bits = LDS offset
#pragma unroll
    for (int j = 0; j < 4; ++j) {
        int cid  = t + (j << 8);                // 16-byte chunk id, 1024 total
        int row  = cid >> 6;                    // 64 chunks per 1KB row
        int col  = (cid & 63) << 2;             // float column
        uint64_t gaddr = (uint64_t)(size_t)(X + (size_t)(m0 + row) * IN_DIM + col);
        uint32_t laddr = lds_base + (uint32_t)cid * 16u;
        asm volatile("global_load_async_to_lds_b128 %0, %1, off"
                     :: "v"(laddr), "v"(gaddr) : "memory");
    }
    asm volatile("s_wait_asynccnt 0" ::: "memory");   // my DMAs landed in LDS
    __syncthreads();                                  // everyone's DMAs visible

    const int wave = t >> 5;
    const int lane = t & 31;
    const int l15  = lane & 15;
    const int hi   = lane >> 4;
    const int n0a  = wave << 5;                 // n-tile 2w
    const int n0b  = n0a + 16;                  // n-tile 2w+1

    const float* browa = W + (size_t)(n0a + l15) * IN_DIM;
    const float* browb = W + (size_t)(n0b + l15) * IN_DIM;
    const float* aLds  = &sA[l15 * IN_DIM];

    v8f acca = {0.f, 0.f, 0.f, 0.f, 0.f, 0.f, 0.f, 0.f};
    v8f accb = {0.f, 0.f, 0.f, 0.f, 0.f, 0.f, 0.f, 0.f};
#pragma unroll 8
    for (int k = 0; k < IN_DIM; k += 4) {
        int off = k + 2 * hi;
        v2f a  = *(const v2f*)(aLds + off);     // ds_load_b64
        v2f ba = *(const v2f*)(browa + off);    // global b64
        v2f bb = *(const v2f*)(browb + off);
        acca = __builtin_amdgcn_wmma_f32_16x16x4_f32(false, a, false, ba,
                                                     (short)0, acca, false, false);
        accb = __builtin_amdgcn_wmma_f32_16x16x4_f32(false, a, false, bb,
                                                     (short)0, accb, false, false);
    }

    float* da = X0 + (size_t)(m0 + 8 * hi) * HC + n0a + l15;
    float* db = X0 + (size_t)(m0 + 8 * hi) * HC + n0b + l15;
#pragma unroll
    for (int v = 0; v < 8; ++v) {
        da[(size_t)v * HC] = acca[v];
        db[(size_t)v * HC] = accb[v];
    }
}

// Scalar tail for N not divisible by 16 (not used for N=50000, kept for safety)
__global__ __launch_bounds__(256) void gemm_tail(const float* __restrict__ X,
                                                 const float* __restrict__ W,
                                                 float* __restrict__ X0,
                                                 int row0, int N) {
    int row = row0 + blockIdx.x;
    if (row >= N) return;
    int n = threadIdx.x;
    const float* xr = X + (size_t)row * IN_DIM;
    const float* wr = W + (size_t)n * IN_DIM;
    float s = 0.f;
    for (int k = 0; k < IN_DIM; ++k) s += xr[k] * wr[k];
    X0[(size_t)row * HC + n] = s;
}

// ---------------------------------------------------------------------------
// Stage 2: scatter-sum X0[vertex] into hyperedges + counts
// ---------------------------------------------------------------------------
__global__ __launch_bounds__(256) void scatter_edge_sum(const float* __restrict__ X0,
                                                        const int* __restrict__ vertex,
                                                        const int* __restrict__ edges,
                                                        float* __restrict__ Xe,
                                                        float* __restrict__ cnt,
                                                        int nnz) {
    int i = blockIdx.x;
    if (i >= nnz) return;
    int c = threadIdx.x;
    int v = vertex[i];
    int e = edges[i];
    atomicAdd(&Xe[(size_t)e * HC + c], X0[(size_t)v * HC + c]);
    if (c == 0) atomicAdd(&cnt[e], 1.0f);
}

// ---------------------------------------------------------------------------
// Stage 3: per-edge mean + attention logits. Block = 256 thr = 8 waves;
// each wave32 owns one head (32 channels) -> shuffle reduction, lane0 writes.
// ---------------------------------------------------------------------------
__global__ __launch_bounds__(256) void edge_mean_alpha(float* __restrict__ Xe,
                                                       const float* __restrict__ cnt,
                                                       const float* __restrict__ att_e,
                                                       float* __restrict__ alpha_e,
                                                       const int* __restrict__ dE) {
    int E = *dE;
    int c = threadIdx.x;
    float ae = att_e[c];                       // [1,H,C] flat == HC
    for (int e = blockIdx.x; e < E; e += gridDim.x) {
        float cn = cnt[e];
        if (cn < 1.f) cn = 1.f;
        float x = Xe[(size_t)e * HC + c] / cn;
        Xe[(size_t)e * HC + c] = x;            // Xe now holds the mean
        float p = x * ae;
#pragma unroll
        for (int off = 16; off > 0; off >>= 1)
            p += __shfl_down(p, off, 32);
        if ((c & 31) == 0)
            alpha_e[(size_t)e * Hh + (c >> 5)] = p;
    }
}

// ---------------------------------------------------------------------------
// Stage 4: segment softmax over destination vertices
// ---------------------------------------------------------------------------
__device__ inline void atomicMaxF(float* addr, float val) {
    int old = __float_as_int(*addr);
    while (__int_as_float(old) < val) {
        int assumed = old;
        old = atomicCAS((int*)addr, assumed, __float_as_int(val));
        if (old == assumed) break;
    }
}

__device__ inline float leaky(float a) { return a > 0.f ? a : NEG_SLOPE * a; }

__global__ void seg_max(const float* __restrict__ alpha_e,
                        const int* __restrict__ vertex,
                        const int* __restrict__ edges,
                        float* __restrict__ amax, int nnzH) {
    int t = blockIdx.x * blockDim.x + threadIdx.x;
    if (t >= nnzH) return;
    int i = t >> 3, h = t & 7;
    float a = leaky(alpha_e[(size_t)edges[i] * Hh + h]);
    atomicMaxF(&amax[(size_t)vertex[i] * Hh + h], a);
}

__global__ void seg_exp_sum(const float* __restrict__ alpha_e,
                            const int* __restrict__ vertex,
                            const int* __restrict__ edges,
                            const float* __restrict__ amax,
                            float* __restrict__ ex,
                            float* __restrict__ denom, int nnzH) {
    int t = blockIdx.x * blockDim.x + threadIdx.x;
    if (t >= nnzH) return;
    int i = t >> 3, h = t & 7;
    int v = vertex[i];
    float a = leaky(alpha_e[(size_t)edges[i] * Hh + h]);
    float m = amax[(size_t)v * Hh + h];
    if (!isfinite(m)) m = 0.f;
    float e = expf(a - m);
    ex[t] = e;                                  // [nnz, H] flat == t
    atomicAdd(&denom[(size_t)v * Hh + h], e);
}

// ---------------------------------------------------------------------------
// Stage 5: attention-weighted scatter back to vertices (accumulate in d_out)
// ---------------------------------------------------------------------------
__global__ __launch_bounds__(256) void scatter_out(const float* __restrict__ Xe,
                                                   const int* __restrict__ vertex,
                                                   const int* __restrict__ edges,
                                                   const float* __restrict__ ex,
                                                   const float* __restrict__ denom,
                                                   float* __restrict__ Xv, int nnz) {
    int i = blockIdx.x;
    if (i >= nnz) return;
    int c = threadIdx.x;
    int h = c >> 5;
    int v = vertex[i];
    int e = edges[i];
    float al = ex[(size_t)i * Hh + h] / (denom[(size_t)v * Hh + h] + 1e-16f);
    atomicAdd(&Xv[(size_t)v * HC + c], Xe[(size_t)e * HC + c] * al);
}

// ---------------------------------------------------------------------------
// Stage 6: in-place row L2 normalize of d_out
// ---------------------------------------------------------------------------
__global__ __launch_bounds__(256) void row_norm(float* __restrict__ Xv, int N) {
    __shared__ float red[8];
    __shared__ float sscale;
    int n = blockIdx.x;
    if (n >= N) return;
    int c = threadIdx.x;
    float x = Xv[(size_t)n * HC + c];
    float s = x * x;
#pragma unroll
    for (int off = 16; off > 0; off >>= 1)
        s += __shfl_down(s, off, 32);
    if ((c & 31) == 0) red[c >> 5] = s;
    __syncthreads();
    if (c == 0) {
        float tt = 0.f;
#pragma unroll
        for (int j = 0; j < 8; ++j) tt += red[j];
        sscale = (tt > 0.f) ? rsqrtf(tt) : 0.f;   // norm>0 ? 1/norm : 0
    }
    __syncthreads();
    Xv[(size_t)n * HC + c] = x * sscale;
}

// ---------------------------------------------------------------------------
extern "C" void kernel_launch(void* const* d_in, const int* in_sizes, int n_in,
                              void* d_out, int out_size, void* d_ws, size_t ws_size,
                              hipStream_t stream) {
    const float* X     = (const float*)d_in[0];
    const float* W     = (const float*)d_in[1];
    const float* att_e = (const float*)d_in[2];
    const int*   vertex = (const int*)d_in[3];
    const int*   edges  = (const int*)d_in[4];
    const int*   dE     = (const int*)d_in[5];
    float* out = (float*)d_out;

    const int N   = in_sizes[0] / IN_DIM;   // 50000
    const int nnz = in_sizes[3];            // 320000

    // workspace layout (floats)
    float* X0      = (float*)d_ws;                       // N*HC
    float* Xe      = X0 + (size_t)N * HC;                // E_CAP*HC
    float* cnt     = Xe + (size_t)E_CAP * HC;            // E_CAP
    float* alpha_e = cnt + E_CAP;                        // E_CAP*H
    float* amax    = alpha_e + (size_t)E_CAP * Hh;       // N*H
    float* denom   = amax + (size_t)N * Hh;              // N*H
    float* ex      = denom + (size_t)N * Hh;             // nnz*H

    // ---- deterministic init every call ----
    {
        size_t n;
        n = (size_t)E_CAP * HC;  fill_f32<<<(n + 255) / 256, 256, 0, stream>>>(Xe, 0.f, n);
        n = (size_t)E_CAP;       fill_f32<<<(n + 255) / 256, 256, 0, stream>>>(cnt, 0.f, n);
        n = (size_t)N * Hh;      fill_f32<<<(n + 255) / 256, 256, 0, stream>>>(amax, -INFINITY, n);
        n = (size_t)N * Hh;      fill_f32<<<(n + 255) / 256, 256, 0, stream>>>(denom, 0.f, n);
        n = (size_t)out_size;    fill_f32<<<(n + 255) / 256, 256, 0, stream>>>(out, 0.f, n);
    }

    // ---- Stage 1: GEMM via WMMA (one block per 16-row M panel) ----
    int mtiles = N / 16;                      // 3125 full panels
    if (mtiles > 0)
        gemm_wmma<<<mtiles, 256, 0, stream>>>(X, W, X0);
    int rem0 = mtiles * 16;
    if (rem0 < N)
        gemm_tail<<<N - rem0, 256, 0, stream>>>(X, W, X0, rem0, N);

    // ---- Stage 2: scatter-sum into hyperedges ----
    scatter_edge_sum<<<nnz, 256, 0, stream>>>(X0, vertex, edges, Xe, cnt, nnz);

    // ---- Stage 3: mean + attention logits (grid-stride over device-read E) ----
    edge_mean_alpha<<<4096, 256, 0, stream>>>(Xe, cnt, att_e, alpha_e, dE);

    // ---- Stage 4: segment softmax ----
    int nnzH = nnz * Hh;
    seg_max<<<(nnzH + 255) / 256, 256, 0, stream>>>(alpha_e, vertex, edges, amax, nnzH);
    seg_exp_sum<<<(nnzH + 255) / 256, 256, 0, stream>>>(alpha_e, vertex, edges, amax, ex, denom, nnzH);

    // ---- Stage 5: weighted scatter back to vertices ----
    scatter_out<<<nnz, 256, 0, stream>>>(Xe, vertex, edges, ex, denom, out, nnz);

    // ---- Stage 6: row L2 normalize ----
    row_norm<<<N, 256, 0, stream>>>(out, N);
}